// Self_Attn_68461778698292
// MI455X (gfx1250) — compile-verified
//
#include <hip/hip_runtime.h>
#include <hip/hip_bf16.h>

// Problem constants (from setup_inputs): B=16, C=64, H=W=64
#define BB 16
#define CC 64
#define NN 4096          // H*W
#define MM 1024          // N/4 after 2x2 pool
#define D8 8             // C/8
#define D2 32            // C/2

typedef float v2f __attribute__((ext_vector_type(2)));
typedef float v8f __attribute__((ext_vector_type(8)));

__device__ __forceinline__ v8f wmma_f32_4(v2f a, v2f b, v8f c) {
    // D(16x16,f32) = A(16x4,f32) x B(4x16,f32) + C
    return __builtin_amdgcn_wmma_f32_16x16x4_f32(false, a, false, b, (short)0, c, false, false);
}

// ---------------------------------------------------------------------------
// Kernel A1: all three 1x1 convolutions at every spatial position.
//   theta -> theta_ws[b][n][8]      (row-contiguous per n for fast tile loads)
//   phi,g (pre-pool) -> tmp_ws[b][o][n], o in [0,40)  (o<8: phi, o>=8: g)
// ---------------------------------------------------------------------------
__global__ void __launch_bounds__(256) proj_all_kernel(
    const float* __restrict__ x,
    const float* __restrict__ w_theta, const float* __restrict__ b_theta,
    const float* __restrict__ w_phi,   const float* __restrict__ b_phi,
    const float* __restrict__ w_g,     const float* __restrict__ b_g,
    float* __restrict__ theta_ws, float* __restrict__ tmp_ws)
{
    __shared__ float wL[48 * 64];
    __shared__ float bL[48];
    const int tid = threadIdx.x;
    for (int i = tid; i < 8 * 64; i += 256)  wL[i]        = w_theta[i];
    for (int i = tid; i < 8 * 64; i += 256)  wL[512 + i]  = w_phi[i];
    for (int i = tid; i < 32 * 64; i += 256) wL[1024 + i] = w_g[i];
    if (tid < 8)        bL[tid] = b_theta[tid];
    else if (tid < 16)  bL[tid] = b_phi[tid - 8];
    else if (tid < 48)  bL[tid] = b_g[tid - 16];
    __syncthreads();

    const int gid = blockIdx.x * 256 + tid;   // grid = B*N/256 = 256 blocks
    const int b = gid >> 12;
    const int n = gid & (NN - 1);

    const float* xp = x + (size_t)b * CC * NN + n;
    float xr[CC];
#pragma unroll
    for (int c = 0; c < CC; ++c) xr[c] = xp[(size_t)c * NN];   // coalesced over n

    for (int o = 0; o < 48; ++o) {
        float acc = bL[o];
#pragma unroll
        for (int c = 0; c < CC; ++c) acc = fmaf(wL[o * 64 + c], xr[c], acc);
        if (o < 8) theta_ws[((size_t)b * NN + n) * 8 + o] = acc;
        else       tmp_ws[(size_t)b * 40 * NN + (size_t)(o - 8) * NN + n] = acc;
    }
}

// ---------------------------------------------------------------------------
// Kernel A2: 2x2 maxpool of phi/g.
//   phi -> phi_ws[b][m][8]   g -> g_ws[b][c][m]   (layouts chosen for kernel B)
// ---------------------------------------------------------------------------
__global__ void __launch_bounds__(256) pool_pg_kernel(
    const float* __restrict__ tmp_ws,
    float* __restrict__ phi_ws, float* __restrict__ g_ws)
{
    const int gid = blockIdx.x * 256 + threadIdx.x;  // grid = B*M/256 = 64 blocks
    const int b = gid >> 10;
    const int m = gid & (MM - 1);
    const int i = m >> 5, j = m & 31;         // pooled coords on 32x32
    const int n0 = i * 128 + j * 2;           // (2i)*64 + 2j
    const float* tp = tmp_ws + (size_t)b * 40 * NN;
    for (int o = 0; o < 40; ++o) {
        const float* p = tp + (size_t)o * NN + n0;
        float v = fmaxf(fmaxf(p[0], p[1]), fmaxf(p[64], p[65]));
        if (o < 8) phi_ws[((size_t)b * MM + m) * 8 + o] = v;
        else       g_ws[(size_t)b * D2 * MM + (size_t)(o - 8) * MM + m] = v;
    }
}

// ---------------------------------------------------------------------------
// Kernel B: fused attention. One workgroup = (batch b, 128 consecutive n).
// Loads phi (M x 8) and g (32 x M) into LDS once, then for 8 tiles of 16 rows:
//   scores via V_WMMA_F32_16X16X4_F32 -> LDS strip (16 x 1024)
//   softmax (VALU + v_exp_f32), normalization deferred
//   PV GEMM via WMMA, K split across the 8 waves, LDS reduction
//   final 1x1 conv w_o + bias + residual, streamed to d_out
// ---------------------------------------------------------------------------
#define SC_STRIDE 1026   // 1024 + 2 dwords pad -> conflict-free b64 A-operand loads
#define G_STRIDE  1026
#define TH_STRIDE 10     // 8 + 2 pad
#define PH_STRIDE 10

__global__ void __launch_bounds__(256) attn_fused_kernel(
    const float* __restrict__ x,
    const float* __restrict__ theta_ws,
    const float* __restrict__ phi_ws,
    const float* __restrict__ g_ws,
    const float* __restrict__ w_o,
    const float* __restrict__ b_o,
    const float* __restrict__ sigma,
    float* __restrict__ outp)
{
    __shared__ __align__(16) float phiL[MM * PH_STRIDE];      // 40 KB
    __shared__ __align__(16) float gL[D2 * G_STRIDE];         // 128.2 KB
    __shared__ __align__(16) float thetaL[16 * TH_STRIDE];
    __shared__ __align__(16) float scoresL[16 * SC_STRIDE];   // 64.1 KB
    __shared__ __align__(16) float redL[8 * 512];             // 16 KB
    __shared__ __align__(16) float w_oL[64 * 33];             // padded stride 33
    __shared__ __align__(16) float attn_gL[D2 * 17];
    __shared__ float invsumL[16];

    const int tid  = threadIdx.x;
    const int wv   = tid >> 5;        // wave id 0..7 (uniform per wave)
    const int lane = tid & 31;
    const int hi   = lane >> 4;       // lane half
    const int lcol = lane & 15;
    const int b      = blockIdx.x >> 5;
    const int chunk  = blockIdx.x & 31;    // 32 chunks of 128 n per batch
    const float sg = sigma[0];

    // ---- stage phi, g, w_o into LDS (coalesced reads, conflict-free writes)
    for (int idx = tid; idx < MM * 8; idx += 256)
        phiL[(idx >> 3) * PH_STRIDE + (idx & 7)] = phi_ws[(size_t)b * MM * 8 + idx];
    for (int idx = tid; idx < D2 * MM; idx += 256)
        gL[(idx >> 10) * G_STRIDE + (idx & 1023)] = g_ws[(size_t)b * D2 * MM + idx];
    for (int idx = tid; idx < 64 * 32; idx += 256)
        w_oL[(idx >> 5) * 33 + (idx & 31)] = w_o[idx];
    __syncthreads();

    const v8f vzero = {0.f, 0.f, 0.f, 0.f, 0.f, 0.f, 0.f, 0.f};

    for (int iter = 0; iter < 8; ++iter) {
        const int n_base = chunk * 128 + iter * 16;

        // ---- load the 16x8 theta tile
        if (tid < 128)
            thetaL[(tid >> 3) * TH_STRIDE + (tid & 7)] =
                theta_ws[((size_t)b * NN + n_base + (tid >> 3)) * 8 + (tid & 7)];
        __syncthreads();

        // ---- phase 1: scores S = theta_tile (16x8) x phi (8xM), WMMA f32 K=4 x2
        {
            v2f a0 = *(const v2f*)&thetaL[lcol * TH_STRIDE + 0 + 2 * hi];
            v2f a1 = *(const v2f*)&thetaL[lcol * TH_STRIDE + 4 + 2 * hi];
            for (int it = 0; it < 8; ++it) {
                const int mt = wv * 8 + it;               // 64 m-tiles over 8 waves
                const float* pb = &phiL[(mt * 16 + lcol) * PH_STRIDE + 2 * hi];
                v2f b0 = *(const v2f*)(pb);
                v2f b1 = *(const v2f*)(pb + 4);
                v8f acc = vzero;
                acc = wmma_f32_4(a0, b0, acc);
                acc = wmma_f32_4(a1, b1, acc);
#pragma unroll
                for (int v = 0; v < 8; ++v)               // D: row=v+8*hi, col=lcol
                    scoresL[(v + 8 * hi) * SC_STRIDE + mt * 16 + lcol] = acc[v];
            }
        }
        __syncthreads();

        // ---- phase 2: softmax over M per row (normalization deferred)
        {
            const int r = tid >> 4;        // row 0..15, 16 threads per row
            const int sub = tid & 15;
            float* row = &scoresL[r * SC_STRIDE];
            float mx = -1e30f;
            for (int j = 0; j < 64; ++j) mx = fmaxf(mx, row[sub + 16 * j]);
#pragma unroll
            for (int off = 8; off >= 1; off >>= 1) mx = fmaxf(mx, __shfl_xor(mx, off));
            float s = 0.f;
            for (int j = 0; j < 64; ++j) {
                float e = __expf(row[sub + 16 * j] - mx);
                row[sub + 16 * j] = e;
                s += e;
            }
#pragma unroll
            for (int off = 8; off >= 1; off >>= 1) s += __shfl_xor(s, off);
            if (sub == 0) invsumL[r] = 1.0f / s;
        }
        __syncthreads();

        // ---- phase 3: attn_g = attn (16xM) x g^T (Mx32); K split over 8 waves
        {
            v8f pacc0 = vzero, pacc1 = vzero;
            const int kbase = wv * 128;
#pragma unroll 4
            for (int ks = 0; ks < 32; ++ks) {
                const int kk = kbase + ks * 4;
                v2f a  = *(const v2f*)&scoresL[lcol * SC_STRIDE + kk + 2 * hi];
                v2f g0 = *(const v2f*)&gL[lcol * G_STRIDE + kk + 2 * hi];
                v2f g1 = *(const v2f*)&gL[(lcol + 16) * G_STRIDE + kk + 2 * hi];
                pacc0 = wmma_f32_4(a, g0, pacc0);
                pacc1 = wmma_f32_4(a, g1, pacc1);
            }
#pragma unroll
            for (int v = 0; v < 8; ++v) {
                redL[wv * 512 +       v * 32 + lane] = pacc0[v];
                redL[wv * 512 + 256 + v * 32 + lane] = pacc1[v];
            }
        }
        __syncthreads();

        // ---- cross-wave reduction + deferred softmax normalization
        for (int rep = 0; rep < 2; ++rep) {
            const int idx = tid + rep * 256;
            float sum = 0.f;
#pragma unroll
            for (int w = 0; w < 8; ++w) sum += redL[w * 512 + idx];
            const int ct = idx >> 8;
            const int rv = (idx >> 5) & 7;
            const int ln = idx & 31;
            const int cch = (ln & 15) + 16 * ct;        // D col = c
            const int nrw = rv + 8 * (ln >> 4);         // D row = n
            attn_gL[cch * 17 + nrw] = sum * invsumL[nrw];
        }
        __syncthreads();

        // ---- phase 5: out = x + sigma * (w_o @ attn_g + b_o)
        {
            const int oo = tid >> 2;            // output channel 0..63
            const int nq = (tid & 3) * 4;       // 4 n-columns per thread
            const float* xp = x    + ((size_t)b * CC + oo) * NN + n_base + nq;
            float*       op = outp + ((size_t)b * CC + oo) * NN + n_base + nq;
            const float bo = b_o[oo];
#pragma unroll
            for (int q = 0; q < 4; ++q) {
                const int nl = nq + q;
                float y = bo;
#pragma unroll
                for (int c2 = 0; c2 < D2; ++c2)
                    y = fmaf(w_oL[oo * 33 + c2], attn_gL[c2 * 17 + nl], y);
                op[q] = xp[q] + sg * y;
            }
        }
        __syncthreads();   // protect thetaL/scoresL before next iteration
    }
}

// ---------------------------------------------------------------------------
extern "C" void kernel_launch(void* const* d_in, const int* in_sizes, int n_in,
                              void* d_out, int out_size, void* d_ws, size_t ws_size,
                              hipStream_t stream) {
    const float* x       = (const float*)d_in[0];
    const float* w_theta = (const float*)d_in[1];
    const float* b_theta = (const float*)d_in[2];
    const float* w_phi   = (const float*)d_in[3];
    const float* b_phi   = (const float*)d_in[4];
    const float* w_g     = (const float*)d_in[5];
    const float* b_g     = (const float*)d_in[6];
    const float* w_o     = (const float*)d_in[7];
    const float* b_o     = (const float*)d_in[8];
    const float* sigma   = (const float*)d_in[9];
    float* out = (float*)d_out;

    float* ws = (float*)d_ws;
    float* theta_ws = ws;                        // 16*4096*8   = 524288 f
    float* tmp_ws   = theta_ws + (size_t)BB * NN * 8;   // 16*40*4096 = 2621440 f
    float* phi_ws   = tmp_ws   + (size_t)BB * 40 * NN;  // 16*1024*8  = 131072 f
    float* g_ws     = phi_ws   + (size_t)BB * MM * 8;   // 16*32*1024 = 524288 f
    // total workspace: 3,801,088 floats = ~14.5 MB

    proj_all_kernel<<<(BB * NN) / 256, 256, 0, stream>>>(
        x, w_theta, b_theta, w_phi, b_phi, w_g, b_g, theta_ws, tmp_ws);
    pool_pg_kernel<<<(BB * MM) / 256, 256, 0, stream>>>(tmp_ws, phi_ws, g_ws);
    attn_fused_kernel<<<BB * 32, 256, 0, stream>>>(
        x, theta_ws, phi_ws, g_ws, w_o, b_o, sigma, out);
}